// Model_55508157334216
// MI455X (gfx1250) — compile-verified
//
#include <hip/hip_runtime.h>

typedef __attribute__((ext_vector_type(2))) float v2f;
typedef __attribute__((ext_vector_type(8))) float v8f;

constexpr int NLAYER = 10;
constexpr int DIMS[NLAYER + 1] = {8, 20, 18, 16, 14, 12, 10, 8, 6, 4, 1};

constexpr int kpad(int k) { return (k + 3) & ~3; }    // K padded to mult of 4
constexpr int npad(int n) { return (n + 15) & ~15; }  // N padded to mult of 16
constexpr int nch(int i) { return kpad(DIMS[i]) / 4; }      // K-chunks (of 4) for layer i
constexpr int nnt(int i) { return npad(DIMS[i + 1]) / 16; } // N-tiles (of 16) for layer i

constexpr int fragOff(int layer) {
  int o = 0;
  for (int i = 0; i < layer; ++i) o += nch(i) * nnt(i);
  return o;
}
constexpr int NFRAG = fragOff(NLAYER);  // 38 B-fragments total

constexpr int biasOff(int layer) {
  int o = 0;
  for (int i = 0; i < layer; ++i) o += nnt(i);
  return o;
}
constexpr int NBIAS = biasOff(NLAYER);  // 12 bias scalars

constexpr int ACT_STRIDE = 33;        // 16x33 f32 activation tile, padded vs bank conflicts
constexpr int WAVES_PER_BLOCK = 8;    // 256 threads, wave32

struct Params {
  const float* W[NLAYER];
  const float* b[NLAYER];
};

// ---- preload all weight fragments (B-matrices, zero padded) + biases into registers ----
template <int I>
__device__ __forceinline__ void preloadB(const Params& p, int ln, int hi, v2f* bf, float* bs) {
  if constexpr (I < NLAYER) {
    constexpr int K = DIMS[I], N = DIMS[I + 1];
    const float* __restrict__ W = p.W[I];
    const float* __restrict__ B = p.b[I];
#pragma unroll
    for (int t = 0; t < nnt(I); ++t) {
      const int n = t * 16 + ln;  // this lane's output column
#pragma unroll
      for (int c = 0; c < nch(I); ++c) {
        // B 4x16 layout: v0 = K {0 | 2} (lanes 0-15 | 16-31), v1 = K {1 | 3}
        const int k0 = c * 4 + hi * 2;
        v2f v;
        v.x = (k0 < K && n < N) ? W[k0 * N + n] : 0.0f;
        v.y = (k0 + 1 < K && n < N) ? W[(k0 + 1) * N + n] : 0.0f;
        bf[fragOff(I) + t * nch(I) + c] = v;
      }
      bs[biasOff(I) + t] = (n < N) ? B[n] : 0.0f;
    }
    preloadB<I + 1>(p, ln, hi, bf, bs);
  }
}

// ---- one layer: D = A*W + bias, activation, scatter D-layout into LDS act tile ----
template <int I>
__device__ __forceinline__ void layerCompute(const v2f* a, const v2f* bf, const float* bs,
                                             float* act, int ln, int hi) {
  constexpr int CH = nch(I), NT = nnt(I);
#pragma unroll
  for (int t = 0; t < NT; ++t) {
    const float bb = bs[biasOff(I) + t];
    v8f acc = {bb, bb, bb, bb, bb, bb, bb, bb};  // C initialized with bias (broadcast per column)
#pragma unroll
    for (int c = 0; c < CH; ++c) {
      acc = __builtin_amdgcn_wmma_f32_16x16x4_f32(
          /*neg_a=*/false, a[c], /*neg_b=*/false, bf[fragOff(I) + t * CH + c],
          /*c_mod=*/(short)0, acc, /*reuse_a=*/false, /*reuse_b=*/false);
    }
#pragma unroll
    for (int g = 0; g < 8; ++g) {
      // D layout: vgpr g holds row M = g + 8*hi, column N = t*16 + ln
      float v = acc[g];
      if constexpr (I == NLAYER - 1) {
        v = 1.0f / (1.0f + __expf(-v));  // sigmoid on final layer
      } else {
        v = fmaxf(v, 0.0f);              // ReLU; padded cols yield relu(0)=0 -> auto zero-pad
      }
      act[(g + hi * 8) * ACT_STRIDE + t * 16 + ln] = v;
    }
  }
}

// ---- layers 1..9: gather A fragments from the LDS act tile, then compute ----
template <int I>
__device__ __forceinline__ void runLayers(const v2f* bf, const float* bs, float* act,
                                          int ln, int hi) {
  if constexpr (I < NLAYER) {
    v2f a[nch(I)];
#pragma unroll
    for (int c = 0; c < nch(I); ++c) {
      // A 16x4 layout: lane ln holds row ln; v0 = K {0 | 2}, v1 = K {1 | 3}
      const int k0 = c * 4 + hi * 2;
      a[c].x = act[ln * ACT_STRIDE + k0];
      a[c].y = act[ln * ACT_STRIDE + k0 + 1];
    }
    layerCompute<I>(a, bf, bs, act, ln, hi);
    runLayers<I + 1>(bf, bs, act, ln, hi);
  }
}

__global__ __launch_bounds__(32 * WAVES_PER_BLOCK) void mlp_wmma_kernel(
    const float* __restrict__ x, Params p, float* __restrict__ out, int nTiles) {
  __shared__ float acts[WAVES_PER_BLOCK][16 * ACT_STRIDE];  // private act tile per wave

  const int lane = threadIdx.x & 31;
  const int wib  = threadIdx.x >> 5;
  const int ln   = lane & 15;
  const int hi   = lane >> 4;
  float* act = &acts[wib][0];

  const int waveId = blockIdx.x * WAVES_PER_BLOCK + wib;
  const int nWaves = (int)gridDim.x * WAVES_PER_BLOCK;

  // Batch-invariant weights live in registers for the whole kernel.
  v2f   bf[NFRAG];
  float bs[NBIAS];
  preloadB<0>(p, ln, hi, bf, bs);

  for (int tile = waveId; tile < nTiles; tile += nWaves) {  // uniform: EXEC stays all-1s for WMMA
    const float* __restrict__ xr = x + (size_t)tile * (16 * 8);
    __builtin_prefetch(xr + (size_t)nWaves * (16 * 8) + (size_t)lane * 4, 0, 3);

    // Layer 0 A fragments straight from global memory (aligned 8B loads, K=8 exactly)
    v2f a0[2];
#pragma unroll
    for (int c = 0; c < 2; ++c) {
      a0[c] = *(const v2f*)(xr + ln * 8 + c * 4 + hi * 2);
    }
    layerCompute<0>(a0, bf, bs, act, ln, hi);
    runLayers<1>(bf, bs, act, ln, hi);

    // Final activations sit in act[row][0]; coalesced 16-float store per tile.
    const float o = act[ln * ACT_STRIDE];
    if (lane < 16) out[tile * 16 + ln] = o;
  }
}

extern "C" void kernel_launch(void* const* d_in, const int* in_sizes, int n_in,
                              void* d_out, int out_size, void* d_ws, size_t ws_size,
                              hipStream_t stream) {
  (void)in_sizes; (void)n_in; (void)d_ws; (void)ws_size;
  const float* x = (const float*)d_in[0];
  Params p;
  for (int i = 0; i < NLAYER; ++i) {
    p.W[i] = (const float*)d_in[1 + 2 * i];
    p.b[i] = (const float*)d_in[2 + 2 * i];
  }
  float* out = (float*)d_out;
  const int nTiles = out_size / 16;  // 65536 tiles of 16 rows
  hipLaunchKernelGGL(mlp_wmma_kernel, dim3(1024), dim3(32 * WAVES_PER_BLOCK), 0, stream,
                     x, p, out, nTiles);
}